// WindowedNorm_40836549050659
// MI455X (gfx1250) — compile-verified
//
#include <hip/hip_runtime.h>

typedef float v2f __attribute__((ext_vector_type(2)));
typedef float v8f __attribute__((ext_vector_type(8)));

namespace {
constexpr int kH = 224;
constexpr int kW = 224;
constexpr int kC = 3;
constexpr int kWin = 47;
constexpr int kPad = 23;      // kWin / 2
constexpr int kTW = 16;       // output columns per workgroup (== WMMA N)
constexpr int kPitch = 17;    // LDS row pitch (floats) to spread banks
constexpr int kHalo = kTW + kWin - 1;   // 62 input columns per row
constexpr int kYTiles = kH / 16;        // 14
constexpr int kTileTasks = kC * kYTiles; // 42 (channel, row-tile) pairs
}

__device__ __forceinline__ int hsIdx(int c, int y, int j) {
    return (c * kH + y) * kPitch + j;
}

__global__ __launch_bounds__(256)
void WindowedNorm_40836549050659_kernel(const float* __restrict__ in,
                                        float* __restrict__ out) {
    extern __shared__ float smem[];
    float* hs  = smem;                      // horizontal window sums of x
    float* hs2 = smem + kC * kH * kPitch;   // horizontal window sums of x*x

    const int b   = blockIdx.y;
    const int x0  = blockIdx.x * kTW;
    const int tid = threadIdx.x;

    // Uniform per-block: does this strip touch the zero-padding in x?
    const bool edge = (x0 < kPad) || (x0 + kTW - 1 + kPad >= kW);

    // ---------------- Phase 1: horizontal sliding box sums ----------------
    // One (row, channel) per thread iteration; 62-wide window in registers,
    // 16 sliding outputs of sum(x) and sum(x^2) to LDS.
    for (int task = tid; task < kH * kC; task += 256) {
        const int y = task / kC;
        const int c = task % kC;
        const long rowBase = (((long)b * kH + y) * kW) * kC + c;
        const float* rp = in + rowBase + (long)(x0 - kPad) * kC;

        float xv[kHalo];
        if (!edge) {
            // Interior strip: single base pointer + immediate offsets.
            #pragma unroll
            for (int i = 0; i < kHalo; ++i) xv[i] = rp[i * kC];
        } else {
            #pragma unroll
            for (int i = 0; i < kHalo; ++i) {
                const int gx = x0 - kPad + i;
                xv[i] = (gx >= 0 && gx < kW) ? rp[i * kC] : 0.0f;
            }
        }
        float s = 0.0f, s2 = 0.0f;
        #pragma unroll
        for (int i = 0; i < kWin; ++i) { s += xv[i]; s2 += xv[i] * xv[i]; }
        hs [hsIdx(c, y, 0)] = s;
        hs2[hsIdx(c, y, 0)] = s2;
        #pragma unroll
        for (int j = 1; j < kTW; ++j) {
            const float aN = xv[j + kWin - 1];
            const float aO = xv[j - 1];
            s  += aN - aO;
            s2 += aN * aN - aO * aO;
            hs [hsIdx(c, y, j)] = s;
            hs2[hsIdx(c, y, j)] = s2;
        }
    }
    __syncthreads();

    // ------ Phase 2: vertical box sums as banded-ones matmul via WMMA ------
    const int lane = tid & 31;
    const int wave = tid >> 5;
    const int M    = lane & 15;   // A-fragment row / B,C,D column
    const int half = lane >> 4;   // 0: K rows {0,1}; 1: K rows {2,3}
    const int col  = M;

    for (int tt = wave; tt < kTileTasks; tt += 8) {
        const int c  = tt / kYTiles;
        const int y0 = (tt % kYTiles) * 16;

        v8f accS  = {0.f, 0.f, 0.f, 0.f, 0.f, 0.f, 0.f, 0.f};
        v8f accS2 = {0.f, 0.f, 0.f, 0.f, 0.f, 0.f, 0.f, 0.f};

        const int yM = y0 + M;
        #pragma unroll 2
        for (int step = 0; step < 16; ++step) {
            const int kk = (y0 - kPad) + step * 4 + half * 2;   // this lane's K rows

            // A fragment: 16x4 banded ones, S[yM][k] = (|yM-k|<=23 && k in range)
            v2f a;
            a.x = (kk     >= 0 && kk     < kH &&
                   (yM - kk)     <= kPad && (kk     - yM) <= kPad) ? 1.0f : 0.0f;
            a.y = (kk + 1 >= 0 && kk + 1 < kH &&
                   (yM - kk - 1) <= kPad && (kk + 1 - yM) <= kPad) ? 1.0f : 0.0f;

            // B fragments from LDS (clamped rows; A==0 kills out-of-range rows)
            const int k0 = kk     < 0 ? 0 : (kk     > kH - 1 ? kH - 1 : kk);
            const int k1 = kk + 1 < 0 ? 0 : (kk + 1 > kH - 1 ? kH - 1 : kk + 1);
            v2f bS, bS2;
            bS.x  = hs [hsIdx(c, k0, col)];
            bS.y  = hs [hsIdx(c, k1, col)];
            bS2.x = hs2[hsIdx(c, k0, col)];
            bS2.y = hs2[hsIdx(c, k1, col)];

            accS  = __builtin_amdgcn_wmma_f32_16x16x4_f32(
                        false, a, false, bS,  (short)0, accS,  false, false);
            accS2 = __builtin_amdgcn_wmma_f32_16x16x4_f32(
                        false, a, false, bS2, (short)0, accS2, false, false);
        }

        // ---------------- Finalize: normalize & store ----------------
        // cf is an exact small integer (<= 47*47), so 1-ulp v_rcp/v_rsq are
        // numerically safe and avoid full IEEE divide expansions.
        const int x    = x0 + col;
        const int cntX = min(x + kPad, kW - 1) - max(x - kPad, 0) + 1;
        #pragma unroll
        for (int r = 0; r < 8; ++r) {
            const int y    = y0 + r + half * 8;           // C/D layout row
            const int cntY = min(y + kPad, kH - 1) - max(y - kPad, 0) + 1;
            const float cf   = (float)(cntX * cntY);
            const float rc   = __builtin_amdgcn_rcpf(cf);
            const float mean = accS[r]  * rc;
            const float q    = accS2[r] * rc;
            float var = fmaf(-mean, mean, q);
            var *= cf * __builtin_amdgcn_rcpf(cf - 1.0f);  // Bessel, edge-aware
            var = fmaxf(var, 0.0f);
            const float sc = __builtin_amdgcn_rsqf(var + 1e-6f);
            const long idx = (((long)b * kH + y) * kW + x) * kC + c;
            out[idx] = (in[idx] - mean) * sc;
        }
    }
}

extern "C" void kernel_launch(void* const* d_in, const int* in_sizes, int n_in,
                              void* d_out, int out_size, void* d_ws, size_t ws_size,
                              hipStream_t stream) {
    (void)n_in; (void)out_size; (void)d_ws; (void)ws_size;
    const float* in = (const float*)d_in[0];
    float* out = (float*)d_out;

    const int B = in_sizes[0] / (kH * kW * kC);
    const size_t smemBytes = (size_t)2 * kC * kH * kPitch * sizeof(float); // 89.25 KB

    (void)hipFuncSetAttribute(
        (const void*)WindowedNorm_40836549050659_kernel,
        hipFuncAttributeMaxDynamicSharedMemorySize, (int)smemBytes);

    dim3 grid(kW / kTW, B);   // 14 column strips x batch
    dim3 block(256);          // 8 wave32s
    WindowedNorm_40836549050659_kernel<<<grid, block, smemBytes, stream>>>(in, out);
}